// NormalizeLayer_44951127720161
// MI455X (gfx1250) — compile-verified
//
#include <hip/hip_runtime.h>

// Native clang vector type -> guarantees global_load_b128 / global_store_b128
typedef __attribute__((ext_vector_type(4))) float v4f;

// out[i] = a[f]*x[i] + b[f], f = i % 128, with a/b derived from 4 groups of 32
// features. Pure streaming: 0.25 FLOP/byte, HBM-bound (~92 us floor @ 23.3 TB/s).
__global__ __launch_bounds__(256) void normalize_stream_kernel(
    const float* __restrict__ x,
    const float* __restrict__ x_min,
    const float* __restrict__ x_max,
    const float* __restrict__ u,
    const float* __restrict__ l,
    float* __restrict__ out,
    long long n)            // total scalar elements
{
    const long long tid    = (long long)blockIdx.x * blockDim.x + threadIdx.x;
    const long long stride = (long long)gridDim.x * blockDim.x;   // multiple of 32
    const long long n4     = n >> 2;                              // float4 count

    // Each thread owns 4 consecutive features: feature = (4*tid) % 128.
    // Group = feature/32 = (tid % 32) / 8. A float4 never crosses a group
    // boundary (4 | 32), and stride % 32 == 0 keeps the group loop-invariant.
    const int g = ((int)(tid & 31)) >> 3;

    // Per-group affine coefficients, computed once per thread.
    const float a = (u[g] - l[g]) / (x_max[g] - x_min[g]);
    const float b = l[g] - x_min[g] * a;

    const v4f* __restrict__ x4 = (const v4f*)x;
    v4f*       __restrict__ o4 = (v4f*)out;

    long long i = tid;

    // Main loop: 4 independent b128 NT loads in flight, then 4 FMAs + NT stores.
    for (; i + 3 * stride < n4; i += 4 * stride) {
        // Speculative prefetch one unroll-window ahead (gfx1250 global_prefetch;
        // OOB translation failures are silently dropped, no counter cost).
        __builtin_prefetch(&x4[i + 4 * stride], 0, 1);

        v4f v0 = __builtin_nontemporal_load(&x4[i]);
        v4f v1 = __builtin_nontemporal_load(&x4[i +     stride]);
        v4f v2 = __builtin_nontemporal_load(&x4[i + 2 * stride]);
        v4f v3 = __builtin_nontemporal_load(&x4[i + 3 * stride]);

        v0 = v0 * a + b;
        v1 = v1 * a + b;
        v2 = v2 * a + b;
        v3 = v3 * a + b;

        __builtin_nontemporal_store(v0, &o4[i]);
        __builtin_nontemporal_store(v1, &o4[i +     stride]);
        __builtin_nontemporal_store(v2, &o4[i + 2 * stride]);
        __builtin_nontemporal_store(v3, &o4[i + 3 * stride]);
    }

    // float4 tail (same group-invariance holds).
    for (; i < n4; i += stride) {
        v4f v = __builtin_nontemporal_load(&x4[i]);
        __builtin_nontemporal_store(v * a + b, &o4[i]);
    }

    // Scalar tail in case n % 4 != 0 (dead for the reference shape: 128 | n).
    for (long long j = (n4 << 2) + tid; j < n; j += stride) {
        const int gg = ((int)(j & 127)) >> 5;
        const float aa = (u[gg] - l[gg]) / (x_max[gg] - x_min[gg]);
        const float bb = l[gg] - x_min[gg] * aa;
        out[j] = x[j] * aa + bb;
    }
}

extern "C" void kernel_launch(void* const* d_in, const int* in_sizes, int n_in,
                              void* d_out, int out_size, void* d_ws, size_t ws_size,
                              hipStream_t stream) {
    const float* x     = (const float*)d_in[0];
    const float* x_min = (const float*)d_in[1];
    const float* x_max = (const float*)d_in[2];
    const float* u     = (const float*)d_in[3];
    const float* l     = (const float*)d_in[4];
    float* out = (float*)d_out;

    const long long n  = (long long)out_size;   // 268,435,456 scalars
    const long long n4 = n >> 2;                // 67,108,864 float4s

    // Target ~16 float4s per thread: 4 unrolled iterations of 4 in-flight b128
    // loads each -> deep memory pipeline, launch overhead amortized.
    const int  tpb   = 256;                     // 8 wave32s per block
    long long  thr   = (n4 + 15) / 16;
    int        blocks = (int)((thr + tpb - 1) / tpb);
    if (blocks < 1) blocks = 1;

    normalize_stream_kernel<<<blocks, tpb, 0, stream>>>(x, x_min, x_max, u, l, out, n);
}